// FastHOGAwareAttention_2413771621019
// MI455X (gfx1250) — compile-verified
//
#include <hip/hip_runtime.h>
#include <hip/hip_bf16.h>
#include <math.h>

// ---------------------------------------------------------------------------
// FastHOGAwareAttention for MI455X (gfx1250), fp32 end-to-end.
// GEMM stages use V_WMMA_F32_16X16X4_F32; GEMM LDS staging uses the Tensor
// Data Mover (TENSOR_LOAD_TO_LDS + s_wait_tensorcnt) when the builtin exists.
// Pipeline (all on `stream`, serialized):
//   1. sobel+HOG      -> hog9 (b,9,N)
//   2. WMMA GEMM      qkv = W_qkv @ x            (576x192x16384 per batch)
//   3. depthwise 3x3  (+ fused hog@W_hog^T into v, + fused q/k sumsq partials)
//   4. tiny partial-reduce -> inverse norms
//   5. WMMA attn partials (48x48, K=16384 chunked, no atomics)
//   6. chunk-reduce + temperature/norm scale + softmax
//   7. M_b = W_proj @ blockdiag(attn_b)          (tiny)
//   8. WMMA GEMM      out = M_b @ v              (192x192x16384 per batch)
// ---------------------------------------------------------------------------

typedef float v2f __attribute__((ext_vector_type(2)));
typedef float v8f __attribute__((ext_vector_type(8)));
typedef unsigned int v4u __attribute__((ext_vector_type(4)));
typedef int v4i __attribute__((ext_vector_type(4)));
typedef int v8i __attribute__((ext_vector_type(8)));

#define BB   8
#define CC   192
#define HH   128
#define WW   128
#define NN   (HH * WW)        // 16384
#define C3   (3 * CC)         // 576
#define ORI  9
#define NH   4
#define HD   48               // CC / NH
#define PI_F 3.14159265358979323846f

// attention K-chunking
#define KC      128
#define CHUNKS  (NN / KC)     // 128
#define BH      (BB * NH)     // 32

#if defined(__has_builtin)
#if __has_builtin(__builtin_amdgcn_tensor_load_to_lds)
#define HAVE_TDM 1
#endif
#if __has_builtin(__builtin_amdgcn_s_wait_tensorcnt)
#define HAVE_TENSORCNT_BUILTIN 1
#endif
#endif

__device__ __forceinline__ void wait_tensorcnt0() {
#if defined(HAVE_TENSORCNT_BUILTIN)
    __builtin_amdgcn_s_wait_tensorcnt(0);
#else
    asm volatile("s_wait_tensorcnt 0x0" ::: "memory");
#endif
}

// ---------------------------------------------------------------------------
// Kernel 1: Sobel (replicate pad, kornia /8 normalization) + HOG binning.
// ---------------------------------------------------------------------------
__global__ void sobel_hog_kernel(const float* __restrict__ x,
                                 float* __restrict__ hog9) {
    __shared__ float patch[18][19];   // +1 col pad vs bank conflicts
    const int tx = threadIdx.x, ty = threadIdx.y;
    const int gx = blockIdx.x * 16 + tx;
    const int gy = blockIdx.y * 16 + ty;
    const int b  = blockIdx.z;
    const int tid = ty * 16 + tx;

    float magsum = 0.f, angsum = 0.f;
    const float* xb = x + (size_t)b * CC * NN;

    for (int c = 0; c < CC; ++c) {
        const float* plane = xb + (size_t)c * NN;
        for (int idx = tid; idx < 18 * 18; idx += 256) {
            int pr = idx / 18, pc = idx % 18;
            int sy = blockIdx.y * 16 - 1 + pr;
            int sx = blockIdx.x * 16 - 1 + pc;
            sy = min(max(sy, 0), HH - 1);
            sx = min(max(sx, 0), WW - 1);
            patch[pr][pc] = plane[sy * WW + sx];
        }
        __syncthreads();
        const int r = ty + 1, cl = tx + 1;
        float dx = (-patch[r-1][cl-1] + patch[r-1][cl+1]
                    - 2.f*patch[r][cl-1] + 2.f*patch[r][cl+1]
                    - patch[r+1][cl-1] + patch[r+1][cl+1]) * 0.125f;
        float dy = (-patch[r-1][cl-1] - 2.f*patch[r-1][cl] - patch[r-1][cl+1]
                    + patch[r+1][cl-1] + 2.f*patch[r+1][cl] + patch[r+1][cl+1]) * 0.125f;
        magsum += sqrtf(dx*dx + dy*dy + 1e-6f);
        angsum += atan2f(dy, dx + 1e-6f);
        __syncthreads();
    }

    const float mag = magsum * (1.f / CC);
    float ang = angsum * (1.f / CC);
    ang = fmodf(ang, PI_F);
    if (ang < 0.f) ang += PI_F;       // python % semantics

    const float bw_inv = (float)ORI / PI_F;
    const float bw = PI_F / (float)ORI;
    const int pix = gy * WW + gx;
    for (int o = 0; o < ORI; ++o) {
        float cen = ((float)o + 0.5f) * bw;
        float w = fmaxf(0.f, 1.f - fabsf(ang - cen) * bw_inv);
        hog9[((size_t)(b * ORI + o)) * NN + pix] = w * mag;
    }
}

// ---------------------------------------------------------------------------
// Generic fp32 WMMA GEMM:  Y[b, m, n] = sum_k A[b?, m, k] * X[b, k, n]
// K fixed at 192. Block = 128 threads (4 waves); wave w owns MT m-tiles.
// X column strip (192 rows x 16 cols) staged in LDS once, via TDM when
// available (tile_dim0=16, tile_dim1=192, tensor_dim0_stride=NN packs rows
// of 16 floats contiguously == xs layout), else cooperative vector loads.
// ---------------------------------------------------------------------------
template <int MT>
__global__ void gemm_wmma_kernel(const float* __restrict__ A, long aBS,
                                 const float* __restrict__ X, long xBS,
                                 float* __restrict__ Y, long yBS) {
    __shared__ float xs[192 * 16];
    const int n0  = blockIdx.x * 16;
    const int b   = blockIdx.y;
    const int tid = threadIdx.x;

    const float* Xb = X + (size_t)b * xBS;

#if defined(HAVE_TDM)
    if (tid < 32) {                       // wave 0 issues the tensor DMA
        unsigned lds_off = (unsigned)(uintptr_t)(&xs[0]);
        unsigned long long ga = (unsigned long long)(uintptr_t)(Xb + n0);
        // D# group0: count=1 | lds_addr | global_addr[56:0] | type=2
        v4u g0 = { 1u, lds_off, (unsigned)ga,
                   ((unsigned)(ga >> 32) & 0x01FFFFFFu) | 0x80000000u };
        // D# group1: data_size=4B, tensor_dim0=NN, tensor_dim1=192,
        //            tile_dim0=16, tile_dim1=192, tensor_dim0_stride=NN
        v8i g1 = { (int)(2u << 16),            // data_size=2 (4 bytes)
                   (int)(16384u << 16),        // tensor_dim0[15:0]
                   (int)(192u << 16),          // tensor_dim0 hi=0 | tensor_dim1 lo
                   (int)(16u << 16),           // tensor_dim1 hi=0 | tile_dim0=16
                   192,                        // tile_dim1=192, tile_dim2=0
                   16384,                      // tensor_dim0_stride lo
                   0, 0 };
        v4i gz = { 0, 0, 0, 0 };
#if __clang_major__ >= 23
        v8i gz8 = { 0, 0, 0, 0, 0, 0, 0, 0 };
        __builtin_amdgcn_tensor_load_to_lds(g0, g1, gz, gz, gz8, 0);
#else
        __builtin_amdgcn_tensor_load_to_lds(g0, g1, gz, gz, 0);
#endif
        wait_tensorcnt0();
    }
    __syncthreads();
#else
    for (int idx = tid; idx < 192 * 16; idx += 128) {
        int k = idx >> 4, nn = idx & 15;
        xs[idx] = Xb[(size_t)k * NN + n0 + nn];
    }
    __syncthreads();
#endif

    const int wave = tid >> 5;
    const int lane = tid & 31;
    const int half = lane >> 4;
    const int l    = lane & 15;

    const float* Ab = A + (size_t)b * aBS;
    float* Yb = Y + (size_t)b * yBS;

    for (int mt = 0; mt < MT; ++mt) {
        const int m0 = (wave * MT + mt) * 16;
        const float* Arow = Ab + (size_t)(m0 + l) * 192;
        v8f acc = {0.f, 0.f, 0.f, 0.f, 0.f, 0.f, 0.f, 0.f};
        #pragma unroll 4
        for (int k0 = 0; k0 < 192; k0 += 4) {
            v2f a, bfrag;
            a.x     = Arow[k0 + 2 * half];
            a.y     = Arow[k0 + 2 * half + 1];
            bfrag.x = xs[(k0 + 2 * half) * 16 + l];
            bfrag.y = xs[(k0 + 2 * half + 1) * 16 + l];
            acc = __builtin_amdgcn_wmma_f32_16x16x4_f32(
                false, a, false, bfrag, (short)0, acc, false, false);
        }
        #pragma unroll
        for (int r = 0; r < 8; ++r)
            Yb[(size_t)(m0 + r + 8 * half) * NN + n0 + l] = acc[r];
    }
}

// ---------------------------------------------------------------------------
// Kernel 3: depthwise 3x3 SAME (zero pad); v channels fuse hog @ W_hog^T;
// q/k channels fuse per-block sum-of-squares partials (saves a 200MB pass).
// ---------------------------------------------------------------------------
__global__ void dwconv_kernel(const float* __restrict__ qkv,
                              const float* __restrict__ W_dw,
                              const float* __restrict__ hog9,
                              const float* __restrict__ W_hog,
                              float* __restrict__ qkvd,
                              float* __restrict__ normpart) {
    __shared__ float red[256];
    const int ch = blockIdx.y;
    const int b  = blockIdx.z;
    const int pix = blockIdx.x * 256 + threadIdx.x;
    const int y = pix >> 7, xx = pix & 127;

    const float* plane = qkv + ((size_t)b * C3 + ch) * NN;
    const float* wk = W_dw + ch * 9;

    float s = 0.f;
    #pragma unroll
    for (int dy = -1; dy <= 1; ++dy) {
        int yy = y + dy;
        if (yy < 0 || yy >= HH) continue;
        #pragma unroll
        for (int dx = -1; dx <= 1; ++dx) {
            int xc = xx + dx;
            if (xc < 0 || xc >= WW) continue;
            s += plane[yy * WW + xc] * wk[(dy + 1) * 3 + dx + 1];
        }
    }
    if (ch >= 2 * CC) {           // v channels: add HOG projection
        const int d = ch - 2 * CC;
        #pragma unroll
        for (int o = 0; o < ORI; ++o)
            s += hog9[((size_t)(b * ORI + o)) * NN + pix] * W_hog[d * ORI + o];
    }
    qkvd[((size_t)b * C3 + ch) * NN + pix] = s;

    if (ch < 2 * CC) {            // q/k channels: partial sum of squares
        red[threadIdx.x] = s * s;
        __syncthreads();
        for (int off = 128; off > 0; off >>= 1) {
            if (threadIdx.x < off) red[threadIdx.x] += red[threadIdx.x + off];
            __syncthreads();
        }
        if (threadIdx.x == 0)
            normpart[((size_t)(b * 2 * CC + ch)) * (NN / 256) + blockIdx.x] = red[0];
    }
}

// ---------------------------------------------------------------------------
// Kernel 4: finish norms from 64 partials per row.
// ---------------------------------------------------------------------------
__global__ void norms_kernel(const float* __restrict__ normpart,
                             float* __restrict__ invq,
                             float* __restrict__ invk) {
    __shared__ float red[64];
    const int c = blockIdx.x;      // 0..383
    const int b = blockIdx.y;
    red[threadIdx.x] = normpart[((size_t)(b * 2 * CC + c)) * 64 + threadIdx.x];
    __syncthreads();
    for (int off = 32; off > 0; off >>= 1) {
        if (threadIdx.x < off) red[threadIdx.x] += red[threadIdx.x + off];
        __syncthreads();
    }
    if (threadIdx.x == 0) {
        float inv = 1.f / fmaxf(sqrtf(red[0]), 1e-12f);
        if (c < CC) invq[b * CC + c] = inv;
        else        invk[b * CC + (c - CC)] = inv;
    }
}

// ---------------------------------------------------------------------------
// Kernel 5: attention partials S_part[chunk][bh][48x48] via WMMA f32 16x16x4.
// 9 waves/block, each owns one 16x16 output tile; K-chunk of 128 staged in LDS
// (row stride 129 -> conflict-free column reads across lanes).
// ---------------------------------------------------------------------------
__global__ void attn_partial_kernel(const float* __restrict__ qkvd,
                                    float* __restrict__ Spart) {
    __shared__ float qs[HD * (KC + 1)];
    __shared__ float ks[HD * (KC + 1)];
    const int chunk = blockIdx.x;
    const int bh = blockIdx.y;
    const int b = bh >> 2, h = bh & 3;
    const int tid = threadIdx.x;
    const int kbase = chunk * KC;

    const float* qbase = qkvd + ((size_t)b * C3 + h * HD) * NN;
    const float* kbase_p = qkvd + ((size_t)b * C3 + CC + h * HD) * NN;
    for (int idx = tid; idx < HD * KC; idx += 288) {
        int r = idx >> 7, kk = idx & (KC - 1);
        qs[r * (KC + 1) + kk] = qbase[(size_t)r * NN + kbase + kk];
        ks[r * (KC + 1) + kk] = kbase_p[(size_t)r * NN + kbase + kk];
    }
    __syncthreads();

    const int wave = tid >> 5;           // 0..8
    const int lane = tid & 31;
    const int half = lane >> 4;
    const int l    = lane & 15;
    const int i0 = (wave / 3) * 16;
    const int j0 = (wave % 3) * 16;

    v8f acc = {0.f, 0.f, 0.f, 0.f, 0.f, 0.f, 0.f, 0.f};
    #pragma unroll 4
    for (int k0 = 0; k0 < KC; k0 += 4) {
        v2f a, bfrag;
        a.x     = qs[(i0 + l) * (KC + 1) + k0 + 2 * half];
        a.y     = qs[(i0 + l) * (KC + 1) + k0 + 2 * half + 1];
        bfrag.x = ks[(j0 + l) * (KC + 1) + k0 + 2 * half];
        bfrag.y = ks[(j0 + l) * (KC + 1) + k0 + 2 * half + 1];
        acc = __builtin_amdgcn_wmma_f32_16x16x4_f32(
            false, a, false, bfrag, (short)0, acc, false, false);
    }
    float* out = Spart + ((size_t)chunk * BH + bh) * (HD * HD);
    #pragma unroll
    for (int r = 0; r < 8; ++r)
        out[(i0 + r + 8 * half) * HD + j0 + l] = acc[r];
}

// ---------------------------------------------------------------------------
// Kernel 6: reduce chunks, scale by temperature * invq_i * invk_j, softmax rows.
// ---------------------------------------------------------------------------
__global__ void softmax_kernel(const float* __restrict__ Spart,
                               const float* __restrict__ invq,
                               const float* __restrict__ invk,
                               const float* __restrict__ temperature,
                               float* __restrict__ attn) {
    __shared__ float srow[HD * HD];
    const int bh = blockIdx.x;
    const int b = bh >> 2, h = bh & 3;
    const int tid = threadIdx.x;
    const float temp = temperature[h];

    for (int idx = tid; idx < HD * HD; idx += 256) {
        float s = 0.f;
        for (int c = 0; c < CHUNKS; ++c)
            s += Spart[((size_t)c * BH + bh) * (HD * HD) + idx];
        int i = idx / HD, j = idx % HD;
        srow[idx] = s * temp * invq[b * CC + h * HD + i] * invk[b * CC + h * HD + j];
    }
    __syncthreads();
    if (tid < HD) {
        const int i = tid;
        float m = -3.402823466e38f;
        for (int j = 0; j < HD; ++j) m = fmaxf(m, srow[i * HD + j]);
        float sum = 0.f;
        float e[HD];
        for (int j = 0; j < HD; ++j) { e[j] = expf(srow[i * HD + j] - m); sum += e[j]; }
        float inv = 1.f / sum;
        for (int j = 0; j < HD; ++j)
            attn[(size_t)bh * (HD * HD) + i * HD + j] = e[j] * inv;
    }
}

// ---------------------------------------------------------------------------
// Kernel 7: M_b = W_proj @ blockdiag(attn_b)   (192x192 per batch, K=48)
// ---------------------------------------------------------------------------
__global__ void mmat_kernel(const float* __restrict__ W_proj,
                            const float* __restrict__ attn,
                            float* __restrict__ Mmat) {
    const int b = blockIdx.y;
    const int idx = blockIdx.x * 256 + threadIdx.x;   // 0..36863
    const int d = idx / CC, cp = idx % CC;
    const int h = cp / HD, j = cp % HD;
    const float* arow = attn + (size_t)(b * NH + h) * (HD * HD);
    float s = 0.f;
    #pragma unroll 8
    for (int i = 0; i < HD; ++i)
        s += W_proj[d * CC + h * HD + i] * arow[i * HD + j];
    Mmat[(size_t)b * CC * CC + idx] = s;
}

// ---------------------------------------------------------------------------
extern "C" void kernel_launch(void* const* d_in, const int* in_sizes, int n_in,
                              void* d_out, int out_size, void* d_ws, size_t ws_size,
                              hipStream_t stream) {
    const float* x      = (const float*)d_in[0];
    const float* W_qkv  = (const float*)d_in[1];
    const float* W_dw   = (const float*)d_in[2];
    const float* W_hog  = (const float*)d_in[3];
    const float* W_proj = (const float*)d_in[4];
    const float* temp   = (const float*)d_in[5];
    float* out = (float*)d_out;

    // workspace layout (floats); total ~650 MB
    float* p = (float*)d_ws;
    float* hog9  = p; p += (size_t)BB * ORI * NN;               // 1.18 M
    float* qkv   = p; p += (size_t)BB * C3 * NN;                // 75.5 M
    float* qkvd  = p; p += (size_t)BB * C3 * NN;                // 75.5 M
    float* invq  = p; p += (size_t)BB * CC;
    float* invk  = p; p += (size_t)BB * CC;
    float* normpart = p; p += (size_t)BB * 2 * CC * (NN / 256); // 0.20 M
    float* Spart = p; p += (size_t)CHUNKS * BH * HD * HD;       // 9.4 M
    float* attn  = p; p += (size_t)BH * HD * HD;
    float* Mmat  = p; p += (size_t)BB * CC * CC;

    // 1. Sobel + HOG
    sobel_hog_kernel<<<dim3(WW / 16, HH / 16, BB), dim3(16, 16), 0, stream>>>(x, hog9);

    // 2. qkv = W_qkv @ x   (M=576 -> 4 waves x 9 m-tiles)
    gemm_wmma_kernel<9><<<dim3(NN / 16, BB), 128, 0, stream>>>(
        W_qkv, 0L, x, (long)CC * NN, qkv, (long)C3 * NN);

    // 3. depthwise 3x3 + fused hog into v + fused q/k sumsq partials
    dwconv_kernel<<<dim3(NN / 256, C3, BB), 256, 0, stream>>>(
        qkv, W_dw, hog9, W_hog, qkvd, normpart);

    // 4. finish norms (64 partials per row)
    norms_kernel<<<dim3(2 * CC, BB), 64, 0, stream>>>(normpart, invq, invk);

    // 5. attention partials (WMMA, K chunked, deterministic)
    attn_partial_kernel<<<dim3(CHUNKS, BH), 288, 0, stream>>>(qkvd, Spart);

    // 6. reduce + scale + softmax
    softmax_kernel<<<dim3(BH), 256, 0, stream>>>(Spart, invq, invk, temp, attn);

    // 7. fused proj-attention matrix
    mmat_kernel<<<dim3(CC * CC / 256, BB), 256, 0, stream>>>(W_proj, attn, Mmat);

    // 8. out = M_b @ v   (M=192 -> 4 waves x 3 m-tiles); v = qkvd channels 384..575
    gemm_wmma_kernel<3><<<dim3(NN / 16, BB), 128, 0, stream>>>(
        Mmat, (long)CC * CC, qkvd + (size_t)2 * CC * NN, (long)C3 * NN,
        out, (long)CC * NN);
}